// SSIMLoss_19052474925668
// MI455X (gfx1250) — compile-verified
//
#include <hip/hip_runtime.h>

// SSIM loss on MI455X (gfx1250): separable 11-tap Gaussian conv expressed as
// banded matmuls on V_WMMA_F32_16X16X4_F32 (full fp32 precision path).

typedef float v2f __attribute__((ext_vector_type(2)));
typedef float v8f __attribute__((ext_vector_type(8)));

#define IMG 512
#define NPIX (32ull * 512ull * 512ull)
#define RPITCH 76   // input region LDS pitch (26 rows valid of 32, 74 cols valid)
#define HPITCH 18   // H scratch pitch (bank-conflict-free for D-frag stores)

__device__ __constant__ float GW[11] = {
    0.00102838f, 0.00759875f, 0.03600077f, 0.10936070f, 0.21300553f,
    0.26601172f,
    0.21300553f, 0.10936070f, 0.03600077f, 0.00759875f, 0.00102838f};

static __device__ __forceinline__ float band(int d) {
  return (d >= 0 && d <= 10) ? GW[d] : 0.0f;
}

static __device__ __forceinline__ v8f wmma4(v2f a, v2f b, v8f c) {
  // D = A(16x4,f32) * B(4x16,f32) + C ; args: neg_a, A, neg_b, B, c_mod, C, reuse_a, reuse_b
  return __builtin_amdgcn_wmma_f32_16x16x4_f32(false, a, false, b, (short)0, c,
                                               false, false);
}

__global__ __launch_bounds__(128) void ssim_tile_kernel(
    const float* __restrict__ pred, const float* __restrict__ tgt,
    double* __restrict__ acc) {
  __shared__ float sP[32 * RPITCH];
  __shared__ float sT[32 * RPITCH];
  __shared__ float sH[4 * 32 * HPITCH];

  const int tid = threadIdx.x;
  const int r0 = blockIdx.y * 16;   // output tile row base
  const int c0 = blockIdx.x * 64;   // workgroup covers 64 output cols (4 waves)
  const size_t base = (size_t)blockIdx.z * (size_t)(IMG * IMG);

  // Cooperative zero-padded halo load: region rows r0-5..r0+20, cols c0-5..c0+68.
  for (int idx = tid; idx < 32 * RPITCH; idx += 128) {
    int rr = idx / RPITCH, cc = idx - rr * RPITCH;
    int gr = r0 - 5 + rr, gc = c0 - 5 + cc;
    float p = 0.f, t = 0.f;
    if (rr < 26 && cc < 74 && gr >= 0 && gr < IMG && gc >= 0 && gc < IMG) {
      size_t off = base + (size_t)gr * IMG + (size_t)gc;
      p = pred[off];
      t = tgt[off];
    }
    sP[idx] = p;
    sT[idx] = t;
  }
  __syncthreads();

  const int wave = tid >> 5;
  const int lane = tid & 31;
  const int lm = lane & 15;
  const int hi = lane >> 4;   // 0 or 1 (lane half)
  const int o0 = hi * 2;      // K sub-offset per A/B fragment layout
  float* Hb = &sH[wave * 32 * HPITCH];
  const int colbase = wave * 16;

  // Banded Gaussian weight fragments: W[j][c] = g[j-c], j in K-dim.
  // Identical per-lane pattern serves as B-operand (H pass) and A-operand (V pass).
  v2f wf[7];
#pragma unroll
  for (int kk = 0; kk < 7; ++kk) {
    int j = kk * 4 + o0;
    wf[kk].x = band(j - lm);
    wf[kk].y = band(j + 1 - lm);
  }

  v8f out[5];
#pragma unroll
  for (int ch = 0; ch < 5; ++ch) out[ch] = v8f{};

#pragma unroll
  for (int ch = 0; ch < 5; ++ch) {
    // ---- horizontal pass: H(32x16) = In(32x28) x Wband(28x16), 2 row blocks
    v8f hlo = v8f{};
    v8f hhi = v8f{};
#pragma unroll
    for (int kk = 0; kk < 7; ++kk) {
      int cix = colbase + kk * 4 + o0;  // even -> 8B aligned LDS read
      v2f pl = *(const v2f*)&sP[lm * RPITCH + cix];
      v2f tl = *(const v2f*)&sT[lm * RPITCH + cix];
      v2f ph = *(const v2f*)&sP[(16 + lm) * RPITCH + cix];
      v2f th = *(const v2f*)&sT[(16 + lm) * RPITCH + cix];
      v2f alo, ahi;
      switch (ch) {
        case 0: alo = pl;       ahi = ph;       break;
        case 1: alo = tl;       ahi = th;       break;
        case 2: alo = pl * pl;  ahi = ph * ph;  break;
        case 3: alo = tl * tl;  ahi = th * th;  break;
        default: alo = pl * tl; ahi = ph * th;  break;
      }
      hlo = wmma4(alo, wf[kk], hlo);
      hhi = wmma4(ahi, wf[kk], hhi);
    }
    // D-fragment layout: VGPR v -> row v + 8*hi, col lm
#pragma unroll
    for (int v = 0; v < 8; ++v) {
      Hb[(v + 8 * hi) * HPITCH + lm] = hlo[v];
      Hb[(16 + v + 8 * hi) * HPITCH + lm] = hhi[v];
    }
    __syncthreads();
    // ---- vertical pass: Out(16x16) = Wband(16x28) x H(28x16)
#pragma unroll
    for (int kk = 0; kk < 7; ++kk) {
      int j = kk * 4 + o0;
      v2f bf;
      bf.x = Hb[j * HPITCH + lm];
      bf.y = Hb[(j + 1) * HPITCH + lm];
      out[ch] = wmma4(wf[kk], bf, out[ch]);
    }
    __syncthreads();
  }

  // ---- elementwise SSIM + reduction
  const float C1 = 0.0001f, C2 = 0.0009f;
  float s = 0.f;
#pragma unroll
  for (int v = 0; v < 8; ++v) {
    float mu_p = out[0][v], mu_t = out[1][v];
    float e_pp = out[2][v], e_tt = out[3][v], e_pt = out[4][v];
    float mu_pp = mu_p * mu_p, mu_tt = mu_t * mu_t, mu_pt = mu_p * mu_t;
    float sig_pp = fmaxf(e_pp - mu_pp, 0.f);
    float sig_tt = fmaxf(e_tt - mu_tt, 0.f);
    float sig_pt = e_pt - mu_pt;
    float num = (2.f * mu_pt + C1) * (2.f * sig_pt + C2);
    float den = (mu_pp + mu_tt + C1) * (sig_pp + sig_tt + C2);
    s += num / (den + 1e-8f);
  }
#pragma unroll
  for (int off = 16; off > 0; off >>= 1) s += __shfl_down(s, off, 32);
  if (lane == 0) atomicAdd(acc, (double)s);
}

__global__ void ssim_zero_acc(double* acc) {
  if (threadIdx.x == 0) acc[0] = 0.0;
}

__global__ void ssim_finalize(const double* __restrict__ acc,
                              float* __restrict__ out) {
  if (threadIdx.x == 0) out[0] = 1.0f - (float)(acc[0] / (double)NPIX);
}

extern "C" void kernel_launch(void* const* d_in, const int* in_sizes, int n_in,
                              void* d_out, int out_size, void* d_ws,
                              size_t ws_size, hipStream_t stream) {
  (void)in_sizes; (void)n_in; (void)out_size; (void)ws_size;
  const float* pred = (const float*)d_in[0];
  const float* tgt = (const float*)d_in[1];
  float* out = (float*)d_out;
  double* acc = (double*)d_ws;

  ssim_zero_acc<<<1, 1, 0, stream>>>(acc);
  dim3 grid(IMG / 64, IMG / 16, 32);  // (8, 32, 32)
  ssim_tile_kernel<<<grid, 128, 0, stream>>>(pred, tgt, acc);
  ssim_finalize<<<1, 1, 0, stream>>>(acc, out);
}